// ChamferPccRateDistortionLoss_11184094838808
// MI455X (gfx1250) — compile-verified
//
#include <hip/hip_runtime.h>

typedef __attribute__((ext_vector_type(2))) float v2f;
typedef __attribute__((ext_vector_type(8))) float v8f;

#define NPTS   4096
#define NBATCH 8
#define NTILES (NPTS / 16)   // 256 tiles of 16 per batch

// One wave = one 16-row strip of one batch. For each of its 16 rows, computes
// min over all 4096 opposing points of squared distance.
//   V_WMMA_F32_16X16X4_F32 with
//     A[M] = [-2x0, -2x1, -2x2, 1]      (loop-invariant)
//     B[N] = [ y0,   y1,   y2,  y^2]    (per column tile)
//     C    = x^2[M] octet               (loop-invariant, SRC2 != VDST)
//   =>  D = x^2 + y^2 - 2 x.y  directly out of the matrix pipe.
// Running min kept with raw v_min3_num_f32 (no NaN-canonicalize, no in-loop
// clamp: max(.,0) is monotone so it is applied once after the loop).
__global__ __launch_bounds__(256)
void chamfer_rowmin_kernel(const float* __restrict__ X,
                           const float* __restrict__ Y,
                           float* __restrict__ out_min) {
    const int lane = threadIdx.x & 31;
    const int wave = threadIdx.x >> 5;
    const int tile = blockIdx.x * 8 + wave;      // global strip id
    const int b    = tile / NTILES;              // batch index
    const int rt   = tile % NTILES;              // row-tile within batch
    const int hi   = lane >> 4;                  // 0 -> K=0,1 ; 1 -> K=2,3
    const int sub  = lane & 15;                  // row/col within 16-tile

    const float* xb = X + (size_t)b * NPTS * 3;
    const float* yb = Y + (size_t)b * NPTS * 3;

    // ---- A tile: rows m = rt*16 + sub ----
    const float* xr = xb + (size_t)(rt * 16 + sub) * 3;
    const float xc0 = xr[0], xc1 = xr[1], xc2 = xr[2];
    const float xnorm = xc0 * xc0 + xc1 * xc1 + xc2 * xc2;

    v2f amat;                                    // K slots: [-2x0,-2x1,-2x2, 1]
    amat.x = hi ? (-2.0f * xc2) : (-2.0f * xc0);
    amat.y = hi ? 1.0f          : (-2.0f * xc1);

    // Loop-invariant C octet: x^2 of the 8 rows M = j + 8*hi this lane covers
    v8f cvec;
#pragma unroll
    for (int j = 0; j < 8; ++j)
        cvec[j] = __shfl(xnorm, j + 8 * hi, 32);

    float rmin[8];
#pragma unroll
    for (int j = 0; j < 8; ++j) rmin[j] = 3.4e38f;

    for (int qt = 0; qt < NTILES; qt += 2) {
        // ---- B tiles qt and qt+1: columns q = qt*16 + sub (+16) ----
        const float* yr0 = yb + (size_t)(qt * 16 + sub) * 3;
        const float* yr1 = yr0 + 16 * 3;
        const float a0 = yr0[0], a1 = yr0[1], a2 = yr0[2];
        const float b0 = yr1[0], b1 = yr1[1], b2 = yr1[2];
        const float an = a0 * a0 + a1 * a1 + a2 * a2;
        const float bn = b0 * b0 + b1 * b1 + b2 * b2;

        v2f bm0, bm1;                            // K slots: [y0, y1, y2, y^2]
        bm0.x = hi ? a2 : a0;
        bm0.y = hi ? an : a1;
        bm1.x = hi ? b2 : b0;
        bm1.y = hi ? bn : b1;

        v8f d0 = __builtin_amdgcn_wmma_f32_16x16x4_f32(
            false, amat, false, bm0, (short)0, cvec, false, false);
        v8f d1 = __builtin_amdgcn_wmma_f32_16x16x4_f32(
            false, amat, false, bm1, (short)0, cvec, false, false);

#pragma unroll
        for (int j = 0; j < 8; ++j) {
            float d0j = d0[j], d1j = d1[j];
            asm("v_min3_num_f32 %0, %0, %1, %2"
                : "+v"(rmin[j])
                : "v"(d0j), "v"(d1j));
        }
    }

    // deferred clamp (monotone under min), then cross-lane min within halves
#pragma unroll
    for (int j = 0; j < 8; ++j) rmin[j] = fmaxf(rmin[j], 0.0f);
#pragma unroll
    for (int m = 1; m <= 8; m <<= 1) {
#pragma unroll
        for (int j = 0; j < 8; ++j)
            rmin[j] = fminf(rmin[j], __shfl_xor(rmin[j], m, 32));
    }

    float* outb = out_min + (size_t)b * NPTS + rt * 16;
    if (sub == 0) {
#pragma unroll
        for (int j = 0; j < 8; ++j)
            outb[hi * 8 + j] = rmin[j];    // hi=0 -> rows 0..7, hi=1 -> rows 8..15
    }
}

// Grid-stride partial sums of -log2(likelihoods); float4 loads, deterministic.
__global__ __launch_bounds__(256)
void bpp_partial_kernel(const float* __restrict__ lik,
                        float* __restrict__ partial, int n) {
    __shared__ float smem[256];
    const float4* lik4 = (const float4*)lik;
    const int n4 = n >> 2;                   // n divisible by 4 here (1572864)
    float s = 0.0f;
    for (int i = blockIdx.x * 256 + threadIdx.x; i < n4; i += gridDim.x * 256) {
        float4 v = lik4[i];
        s -= __log2f(v.x) + __log2f(v.y) + __log2f(v.z) + __log2f(v.w);
    }
    for (int i = (n4 << 2) + blockIdx.x * 256 + threadIdx.x; i < n;
         i += gridDim.x * 256)
        s -= __log2f(lik[i]);
    smem[threadIdx.x] = s;
    __syncthreads();
    for (int w = 128; w > 0; w >>= 1) {
        if ((int)threadIdx.x < w) smem[threadIdx.x] += smem[threadIdx.x + w];
        __syncthreads();
    }
    if (threadIdx.x == 0) partial[blockIdx.x] = smem[0];
}

__global__ __launch_bounds__(256)
void finalize_kernel(const float* __restrict__ rowmin,
                     const float* __restrict__ colmin,
                     const float* __restrict__ bppPartial,
                     int nMin, int nPart, float* __restrict__ out) {
    __shared__ float smemR[256];
    __shared__ float smemB[256];
    float sr = 0.0f;
    for (int i = threadIdx.x; i < nMin; i += 256) sr += rowmin[i] + colmin[i];
    float sb = 0.0f;
    for (int i = threadIdx.x; i < nPart; i += 256) sb += bppPartial[i];
    smemR[threadIdx.x] = sr;
    smemB[threadIdx.x] = sb;
    __syncthreads();
    for (int w = 128; w > 0; w >>= 1) {
        if ((int)threadIdx.x < w) {
            smemR[threadIdx.x] += smemR[threadIdx.x + w];
            smemB[threadIdx.x] += smemB[threadIdx.x + w];
        }
        __syncthreads();
    }
    if (threadIdx.x == 0) {
        const float inv = 1.0f / (float)(NBATCH * NPTS);   // 1/32768
        float rec = smemR[0] * inv;   // sum(rowmin)+sum(colmin) over B*P
        float bpp = smemB[0] * inv;
        out[0] = rec + bpp;  // loss
        out[1] = bpp;        // bpp_loss
        out[2] = rec;        // rec_loss
    }
}

extern "C" void kernel_launch(void* const* d_in, const int* in_sizes, int n_in,
                              void* d_out, int out_size, void* d_ws, size_t ws_size,
                              hipStream_t stream) {
    const float* x_hat = (const float*)d_in[0];
    const float* pos   = (const float*)d_in[1];
    const float* lik   = (const float*)d_in[2];
    float* out = (float*)d_out;

    float* ws      = (float*)d_ws;
    float* rowmin  = ws;                     // 32768 floats (x_hat -> pos mins)
    float* colmin  = ws + NBATCH * NPTS;     // 32768 floats (pos -> x_hat mins)
    float* bppPart = ws + 2 * NBATCH * NPTS; // 256 floats

    const int nLik = in_sizes[2];
    const int nStrips = NBATCH * NTILES;     // 2048 waves total
    dim3 blk(256);
    dim3 grdCham(nStrips / 8);               // 8 waves per block

    hipLaunchKernelGGL(chamfer_rowmin_kernel, grdCham, blk, 0, stream,
                       x_hat, pos, rowmin);
    hipLaunchKernelGGL(chamfer_rowmin_kernel, grdCham, blk, 0, stream,
                       pos, x_hat, colmin);
    hipLaunchKernelGGL(bpp_partial_kernel, dim3(256), blk, 0, stream,
                       lik, bppPart, nLik);
    hipLaunchKernelGGL(finalize_kernel, dim3(1), blk, 0, stream,
                       rowmin, colmin, bppPart, NBATCH * NPTS, 256, out);
}